// GlobalFilter_free_size_21981642621512
// MI455X (gfx1250) — compile-verified
//
#include <hip/hip_runtime.h>

// ---------------- CDNA5 WMMA types ----------------
typedef __attribute__((ext_vector_type(2))) float v2f;
typedef __attribute__((ext_vector_type(8))) float v8f;

__device__ __forceinline__ v8f wmma4(v2f a, v2f b, v8f c) {
  // D(16x16,f32) = A(16x4,f32) * B(4x16,f32) + C
  return __builtin_amdgcn_wmma_f32_16x16x4_f32(false, a, false, b, (short)0, c,
                                               false, false);
}

// Optional gfx1250 async global->LDS path (ASYNCcnt-tracked), guarded so the
// file compiles on toolchains that don't expose the builtins.
#if defined(__has_builtin)
#if __has_builtin(__builtin_amdgcn_global_load_async_to_lds_b32) && \
    __has_builtin(__builtin_amdgcn_s_wait_asynccnt)
#define USE_ASYNC_LDS 1
#endif
#endif

#ifdef USE_ASYNC_LDS
// clang declares the builtin's pointer params as int* in AS1 (prints as
// "__device__ int *") and AS3 ("__shared__ int *").
typedef __attribute__((address_space(1))) int as1_int;
typedef __attribute__((address_space(3))) int as3_int;
#endif

#define HH 256
#define WW 256
#define WF 129
#define CCH 64
#define NPOS 33024  // HH*WF
#define SPITCH 130  // LDS pitch for half-spectrum rows (bank-conflict free)
#define XPITCH 260  // LDS pitch for x row tile

// ---------------- K0: DFT matrix D[n][k] = exp(-2*pi*i*n*k/256) ------------
__global__ void k0_dft(float* __restrict__ Dre, float* __restrict__ Dim) {
  int idx = blockIdx.x * 256 + threadIdx.x;  // 0..65535
  int n = idx >> 8, k = idx & 255;
  int ph = (n * k) & 255;
  float ang = (float)ph * 0.024543692606170259f;  // 2*pi/256
  float s, c;
  sincosf(ang, &s, &c);
  Dre[idx] = c;
  Dim[idx] = -s;
}

// ---------------- K1: conv1x1(5->64) + InstanceNorm + ReLU ------------------
__global__ void k1_layer1(const float* __restrict__ w1,
                          const float* __restrict__ b1,
                          float* __restrict__ h1) {
  const int c = blockIdx.x;  // 0..63
  const int t = threadIdx.x;
  __shared__ float rs[256], rq[256];
  const float a0 = w1[c * 5 + 0], a1 = w1[c * 5 + 1], a2 = w1[c * 5 + 2];
  const float a3 = w1[c * 5 + 3], a4 = w1[c * 5 + 4], bb = b1[c];
  float sum = 0.f, sq = 0.f;
  for (int p = t; p < NPOS; p += 256) {
    float gx = (float)(p / WF), gy = (float)(p % WF);
    float v = a0 * gx + a1 * gy + a2 * gx * gx + a3 * gy * gy + a4 * gy * gx + bb;
    sum += v; sq += v * v;
  }
  rs[t] = sum; rq[t] = sq;
  __syncthreads();
  for (int o = 128; o > 0; o >>= 1) {
    if (t < o) { rs[t] += rs[t + o]; rq[t] += rq[t + o]; }
    __syncthreads();
  }
  const float mean = rs[0] * (1.f / NPOS);
  const float var = rq[0] * (1.f / NPOS) - mean * mean;
  const float inv = rsqrtf(var + 1e-5f);
  for (int p = t; p < NPOS; p += 256) {
    float gx = (float)(p / WF), gy = (float)(p % WF);
    float v = a0 * gx + a1 * gy + a2 * gx * gx + a3 * gy * gy + a4 * gy * gx + bb;
    h1[(size_t)p * CCH + c] = fmaxf((v - mean) * inv, 0.f);
  }
}

// ---------------- K2: conv1x1(64->128) + InstanceNorm -> 0.02*(re,im) -------
__global__ void k2_layer2(const float* __restrict__ h1,
                          const float* __restrict__ w2,
                          const float* __restrict__ b2,
                          float* __restrict__ wre, float* __restrict__ wim) {
  const int d = blockIdx.x;  // 0..127
  const int t = threadIdx.x;
  __shared__ float sw[CCH];
  __shared__ float rs[256], rq[256];
  if (t < CCH) sw[t] = w2[d * CCH + t];
  __syncthreads();
  const float bb = b2[d];
  float sum = 0.f, sq = 0.f;
  for (int p = t; p < NPOS; p += 256) {
    const float* hp = h1 + (size_t)p * CCH;
    float v = bb;
#pragma unroll
    for (int cc = 0; cc < CCH; ++cc) v += hp[cc] * sw[cc];
    sum += v; sq += v * v;
  }
  rs[t] = sum; rq[t] = sq;
  __syncthreads();
  for (int o = 128; o > 0; o >>= 1) {
    if (t < o) { rs[t] += rs[t + o]; rq[t] += rq[t + o]; }
    __syncthreads();
  }
  const float mean = rs[0] * (1.f / NPOS);
  const float var = rq[0] * (1.f / NPOS) - mean * mean;
  const float inv = rsqrtf(var + 1e-5f);
  float* dst = (d < CCH) ? (wre + (size_t)d * NPOS)
                         : (wim + (size_t)(d - CCH) * NPOS);
  for (int p = t; p < NPOS; p += 256) {
    const float* hp = h1 + (size_t)p * CCH;
    float v = bb;
#pragma unroll
    for (int cc = 0; cc < CCH; ++cc) v += hp[cc] * sw[cc];
    dst[p] = 0.02f * (v - mean) * inv;
  }
}

// ---------------- Fused per-image rfft2 -> (1+w) -> irfft2 ------------------
// One workgroup per image (b,c). Half-spectrum lives entirely in LDS.
// LDS layout (floats):
//   sRe [256][130]  @ 0
//   sIm [256][130]  @ 33280
//   tRe [256][16]   @ 66560   (column-block scratch, complex)
//   tIm [256][16]   @ 70656
//   xb  [16][260]   @ 74752   (one 16-row tile of x)
// total = 78912 floats = 315,648 B  (<= 320KB/WGP)
__global__ void __launch_bounds__(256, 1)
gfilter_fused(const float* __restrict__ x, const float* __restrict__ Dre,
              const float* __restrict__ Dim, const float* __restrict__ wre,
              const float* __restrict__ wim, float* __restrict__ out) {
  extern __shared__ float lds[];
  float* sRe = lds;
  float* sIm = lds + 33280;
  float* tRe = lds + 66560;
  float* tIm = lds + 70656;
  float* xb = lds + 74752;

  const int img = blockIdx.x;  // 0..511 == b*C + c
  const int c = img & (CCH - 1);
  const float* xim = x + (size_t)img * (HH * WW);
  float* oim = out + (size_t)img * (HH * WW);

  const int t = threadIdx.x;   // 0..255 (8 waves)
  const int wv = t >> 5;       // wave id 0..7
  const int lane = t & 31;
  const int lm = lane & 15;    // row/col index within 16
  const int lh = lane >> 4;    // half select

  __builtin_prefetch(Dre, 0, 3);
  __builtin_prefetch(Dim, 0, 3);

  // ====================== Stage 1: row rFFT ======================
  // S[r][k] = sum_n x[r][n] * D[n][k],  k = 0..128 kept (Hermitian half)
  for (int rb = 0; rb < 16; ++rb) {
    __syncthreads();  // protect xb against previous iteration readers
#ifdef USE_ASYNC_LDS
    for (int i = t; i < 4096; i += 256) {
      float* gp = const_cast<float*>(xim) + rb * 4096 + i;
      float* lp = xb + (i >> 8) * XPITCH + (i & 255);
      __builtin_amdgcn_global_load_async_to_lds_b32(
          (as1_int*)gp, (as3_int*)lp, 0, 0);
    }
    __builtin_amdgcn_s_wait_asynccnt(0);
#else
    for (int i = t; i < 4096; i += 256)
      xb[(i >> 8) * XPITCH + (i & 255)] = xim[rb * 4096 + i];
#endif
    __syncthreads();
    for (int kt = wv; kt < 9; kt += 8) {  // 9 k-tiles cover k=0..143
      v8f ar = {}, ai = {};
#pragma unroll 4
      for (int ks = 0; ks < 64; ++ks) {
        const int k0 = 4 * ks + 2 * lh;  // K index base (n)
        v2f a = {xb[lm * XPITCH + k0], xb[lm * XPITCH + k0 + 1]};
        v2f br = {Dre[k0 * 256 + kt * 16 + lm], Dre[(k0 + 1) * 256 + kt * 16 + lm]};
        v2f bi = {Dim[k0 * 256 + kt * 16 + lm], Dim[(k0 + 1) * 256 + kt * 16 + lm]};
        ar = wmma4(a, br, ar);
        ai = wmma4(a, bi, ai);
      }
#pragma unroll
      for (int j = 0; j < 8; ++j) {
        const int m = j + 8 * lh;      // row within block
        const int k = kt * 16 + lm;    // frequency
        if (k < WF) {
          sRe[(rb * 16 + m) * SPITCH + k] = ar[j];
          sIm[(rb * 16 + m) * SPITCH + k] = ai[j];
        }
      }
    }
  }

  // ============ Stage 2: column FFT -> filter -> column IFFT ============
  for (int cb = 0; cb < 9; ++cb) {  // 16-column blocks (last covers only col 128)
    const int colbase = cb * 16;
    __syncthreads();
    // col FFT: T[kh][col] = sum_r S[r][col] * D[r][kh]
    // Each wave owns kh-tiles {wv, wv+8}; B (spectrum columns) shared.
    {
      const int m0 = wv * 16, m1 = (wv + 8) * 16;
      v8f cr0 = {}, ci0 = {}, cr1 = {}, ci1 = {};
#pragma unroll 2
      for (int ks = 0; ks < 64; ++ks) {
        const int k0 = 4 * ks + 2 * lh;  // r
        v2f br = {sRe[k0 * SPITCH + colbase + lm], sRe[(k0 + 1) * SPITCH + colbase + lm]};
        v2f bi = {sIm[k0 * SPITCH + colbase + lm], sIm[(k0 + 1) * SPITCH + colbase + lm]};
        v2f re0 = {Dre[k0 * 256 + m0 + lm], Dre[(k0 + 1) * 256 + m0 + lm]};
        v2f di0 = {Dim[k0 * 256 + m0 + lm], Dim[(k0 + 1) * 256 + m0 + lm]};
        v2f re1 = {Dre[k0 * 256 + m1 + lm], Dre[(k0 + 1) * 256 + m1 + lm]};
        v2f di1 = {Dim[k0 * 256 + m1 + lm], Dim[(k0 + 1) * 256 + m1 + lm]};
        cr0 = wmma4(re0, br, cr0);
        cr0 = wmma4(-di0, bi, cr0);  // f32 WMMA has no A-neg: negate operand
        ci0 = wmma4(re0, bi, ci0);
        ci0 = wmma4(di0, br, ci0);
        cr1 = wmma4(re1, br, cr1);
        cr1 = wmma4(-di1, bi, cr1);
        ci1 = wmma4(re1, bi, ci1);
        ci1 = wmma4(di1, br, ci1);
      }
#pragma unroll
      for (int j = 0; j < 8; ++j) {
        const int r0 = m0 + j + 8 * lh;
        const int r1 = m1 + j + 8 * lh;
        tRe[r0 * 16 + lm] = cr0[j];
        tIm[r0 * 16 + lm] = ci0[j];
        tRe[r1 * 16 + lm] = cr1[j];
        tIm[r1 * 16 + lm] = ci1[j];
      }
    }
    __syncthreads();
    // filter: X *= (1 + w[c,kh,kw])
    for (int i = t; i < 4096; i += 256) {
      const int kh = i >> 4, col = i & 15;
      const int kw = colbase + col;
      if (kw < WF) {
        const size_t wi = (size_t)c * NPOS + kh * WF + kw;
        const float wr = 1.0f + wre[wi];
        const float wii = wim[wi];
        const float xr = tRe[i], xi = tIm[i];
        tRe[i] = xr * wr - xi * wii;
        tIm[i] = xr * wii + xi * wr;
      }
    }
    __syncthreads();
    // col IFFT: S[r][col] = sum_kh T[kh][col] * conj(D)[kh][r]
    // Each wave owns r-tiles {wv, wv+8}; B (filtered columns) shared.
    {
      const int m0 = wv * 16, m1 = (wv + 8) * 16;
      v8f cr0 = {}, ci0 = {}, cr1 = {}, ci1 = {};
#pragma unroll 2
      for (int ks = 0; ks < 64; ++ks) {
        const int k0 = 4 * ks + 2 * lh;  // kh
        v2f br = {tRe[k0 * 16 + lm], tRe[(k0 + 1) * 16 + lm]};
        v2f bi = {tIm[k0 * 16 + lm], tIm[(k0 + 1) * 16 + lm]};
        v2f re0 = {Dre[k0 * 256 + m0 + lm], Dre[(k0 + 1) * 256 + m0 + lm]};
        v2f di0 = {Dim[k0 * 256 + m0 + lm], Dim[(k0 + 1) * 256 + m0 + lm]};
        v2f re1 = {Dre[k0 * 256 + m1 + lm], Dre[(k0 + 1) * 256 + m1 + lm]};
        v2f di1 = {Dim[k0 * 256 + m1 + lm], Dim[(k0 + 1) * 256 + m1 + lm]};
        cr0 = wmma4(re0, br, cr0);
        cr0 = wmma4(di0, bi, cr0);   // conj(D): + Dim*Bi
        ci0 = wmma4(re0, bi, ci0);
        ci0 = wmma4(-di0, br, ci0);  // conj(D): Ai = -Dim
        cr1 = wmma4(re1, br, cr1);
        cr1 = wmma4(di1, bi, cr1);
        ci1 = wmma4(re1, bi, ci1);
        ci1 = wmma4(-di1, br, ci1);
      }
      const int kw = colbase + lm;
      if (kw < WF) {
#pragma unroll
        for (int j = 0; j < 8; ++j) {
          const int r0 = m0 + j + 8 * lh;
          const int r1 = m1 + j + 8 * lh;
          sRe[r0 * SPITCH + kw] = cr0[j];
          sIm[r0 * SPITCH + kw] = ci0[j];
          sRe[r1 * SPITCH + kw] = cr1[j];
          sIm[r1 * SPITCH + kw] = ci1[j];
        }
      }
    }
  }
  __syncthreads();

  // ====================== Stage 3: row irFFT ======================
  // out[r][n] = (1/65536) * sum_k ( Sf_re[r][k]*Dre[k][n] + Sf_im[r][k]*Dim[k][n] )
  // with Hermitian extension Sf[k>128] = conj(S[256-k]).
  // Pair adjacent output-column tiles so the gathered A operand is shared.
  for (int pj = wv; pj < 128; pj += 8) {
    const int rb = pj >> 3;            // row block 0..15
    const int nt0 = (pj & 7) * 2;      // output col tiles nt0, nt0+1
    const int n0 = nt0 * 16, n1 = n0 + 16;
    v8f acc0 = {}, acc1 = {};
#pragma unroll 2
    for (int ks = 0; ks < 64; ++ks) {
      const int k0 = 4 * ks + 2 * lh;
      const int ka = (k0 <= 128) ? k0 : 256 - k0;
      const int kb = (k0 + 1 <= 128) ? k0 + 1 : 255 - k0;
      const float sa = (k0 <= 128) ? 1.f : -1.f;
      const float sb = (k0 + 1 <= 128) ? 1.f : -1.f;
      const int row = rb * 16 + lm;
      v2f arr = {sRe[row * SPITCH + ka], sRe[row * SPITCH + kb]};
      v2f aii = {sa * sIm[row * SPITCH + ka], sb * sIm[row * SPITCH + kb]};
      v2f br0 = {Dre[k0 * 256 + n0 + lm], Dre[(k0 + 1) * 256 + n0 + lm]};
      v2f bi0 = {Dim[k0 * 256 + n0 + lm], Dim[(k0 + 1) * 256 + n0 + lm]};
      v2f br1 = {Dre[k0 * 256 + n1 + lm], Dre[(k0 + 1) * 256 + n1 + lm]};
      v2f bi1 = {Dim[k0 * 256 + n1 + lm], Dim[(k0 + 1) * 256 + n1 + lm]};
      acc0 = wmma4(arr, br0, acc0);
      acc0 = wmma4(aii, bi0, acc0);
      acc1 = wmma4(arr, br1, acc1);
      acc1 = wmma4(aii, bi1, acc1);
    }
    const float s = 1.0f / 65536.0f;  // combined ortho fwd+inv normalization
#pragma unroll
    for (int j = 0; j < 8; ++j) {
      const int r = rb * 16 + j + 8 * lh;
      oim[r * 256 + n0 + lm] = acc0[j] * s;
      oim[r * 256 + n1 + lm] = acc1[j] * s;
    }
  }
}

// ---------------- launcher ----------------
extern "C" void kernel_launch(void* const* d_in, const int* in_sizes, int n_in,
                              void* d_out, int out_size, void* d_ws,
                              size_t ws_size, hipStream_t stream) {
  const float* x = (const float*)d_in[0];
  const float* w1 = (const float*)d_in[1];
  const float* b1 = (const float*)d_in[2];
  const float* w2 = (const float*)d_in[3];
  const float* b2 = (const float*)d_in[4];
  float* out = (float*)d_out;

  char* ws = (char*)d_ws;
  float* Dre = (float*)(ws);                       // 256KB
  float* Dim = (float*)(ws + 262144);              // 256KB
  float* h1 = (float*)(ws + 524288);               // 33024*64*4 = 8.45MB
  float* wre = (float*)(ws + 524288 + 8454144);    // 64*33024*4
  float* wim = (float*)(ws + 524288 + 2 * 8454144);

  k0_dft<<<256, 256, 0, stream>>>(Dre, Dim);
  k1_layer1<<<64, 256, 0, stream>>>(w1, b1, h1);
  k2_layer2<<<128, 256, 0, stream>>>(h1, w2, b2, wre, wim);

  const size_t smem = 78912 * sizeof(float);  // 315,648 B <= 320KB/WGP
  gfilter_fused<<<512, 256, smem, stream>>>(x, Dre, Dim, wre, wim, out);
}